// PermuteSmoothedDistillLoss_45414984188280
// MI455X (gfx1250) — compile-verified
//
#include <hip/hip_runtime.h>
#include <hip/hip_bf16.h>

// ---------------------------------------------------------------------------
// PermuteSmoothedDistillLoss for MI455X (gfx1250, wave32).
//
// Memory-bound: only logits/bias/teacher_probs/labels are read (40 B/row,
// ~5.2 MB total); the 402 MB `hidden` tensor is unused by the reference and
// deliberately never touched. Two-pass deterministic reduction:
//   kernel 1: 1 row/thread, full per-row math, wave shuffle + LDS block
//             reduce -> one f32 partial per block in d_ws.
//   kernel 2: single wave32; lane-strided sum of partials, then a
//             V_WMMA_F32_16X16X4_F32 with an all-ones B matrix performs the
//             32-lane cross-lane reduction (A 16x4 layout: lanes 0-15 feed
//             K=0, lanes 16-31 feed K=2, so D[m][*] = v_m + v_{m+16});
//             per-lane sum of the 8 D VGPRs + one shfl_xor(16) finishes.
// ---------------------------------------------------------------------------

typedef float v2f __attribute__((ext_vector_type(2)));
typedef float v8f __attribute__((ext_vector_type(8)));

#define NC 3

__global__ void __launch_bounds__(256)
psd_loss_rows_kernel(const float* __restrict__ logits,
                     const float* __restrict__ bias,
                     const float* __restrict__ teacher,
                     const int*   __restrict__ labels,
                     float* __restrict__ partials,
                     int nrows) {
    const int row = blockIdx.x * blockDim.x + threadIdx.x;

    float local = 0.0f;
    if (row < nrows) {
        const float* lg = logits  + NC * row;
        const float* tp = teacher + NC * row;

        // --- log_softmax of student logits (stable) ---
        const float l0 = lg[0], l1 = lg[1], l2 = lg[2];
        const float m  = fmaxf(l0, fmaxf(l1, l2));
        const float z  = expf(l0 - m) + expf(l1 - m) + expf(l2 - m);
        const float lse = m + logf(z);
        const float lp0 = l0 - lse, lp1 = l1 - lse, lp2 = l2 - lse;

        // --- w = 1 - exp(bias[label]) ---
        const int lab = labels[row];
        const float w = 1.0f - expf(bias[NC * row + lab]);

        // --- power-weighted teacher renorm: t^w / sum(t^w) ---
        const float t0 = tp[0], t1 = tp[1], t2 = tp[2];
        const float x0 = expf(w * logf(t0));
        const float x1 = expf(w * logf(t1));
        const float x2 = expf(w * logf(t2));
        const float inv = 1.0f / (x0 + x1 + x2);
        const float n0 = x0 * inv, n1 = x1 * inv, n2 = x2 * inv;

        // --- argmax (first-max semantics, matches jnp.argmax) ---
        int a = 0; float best = n0;
        if (n1 > best) { a = 1; best = n1; }
        if (n2 > best) { a = 2; }

        // --- swap the two non-argmax entries: perm[c] = c==a ? c : 3-a-c ---
        float s0, s1, s2;
        if (a == 0)      { s0 = n0; s1 = n2; s2 = n1; }
        else if (a == 1) { s0 = n2; s1 = n1; s2 = n0; }
        else             { s0 = n1; s1 = n0; s2 = n2; }

        local = -(s0 * lp0 + s1 * lp1 + s2 * lp2);
    }

    // --- wave32 shuffle reduction ---
    #pragma unroll
    for (int off = 16; off > 0; off >>= 1)
        local += __shfl_xor(local, off, 32);

    __shared__ float smem[8];  // 256 threads = 8 waves
    const int wid = threadIdx.x >> 5;
    const int lid = threadIdx.x & 31;
    if (lid == 0) smem[wid] = local;
    __syncthreads();

    if (threadIdx.x == 0) {
        float s = 0.0f;
        #pragma unroll
        for (int i = 0; i < 8; ++i) s += smem[i];
        partials[blockIdx.x] = s;
    }
}

// Single wave (32 threads). Final deterministic reduction + mean.
// Uses V_WMMA_F32_16X16X4_F32 with an all-ones B matrix to sum across lanes.
__global__ void __launch_bounds__(32)
psd_loss_reduce_kernel(const float* __restrict__ partials,
                       int nparts,
                       float* __restrict__ out,
                       float inv_b) {
    const int lane = threadIdx.x;  // 0..31, EXEC all ones

    float v = 0.0f;
    for (int i = lane; i < nparts; i += 32)
        v += partials[i];
    // (divergence on last loop iter reconverges here; EXEC all-1s for WMMA)

    // A: 16x4 f32 -> 2 VGPRs. Lanes 0-15 supply K=0 (VGPR0), K=1 (VGPR1);
    // lanes 16-31 supply K=2, K=3. Put v in VGPR0, 0 in VGPR1 so
    // A[m][0] = v_m, A[m][2] = v_{m+16}, other columns zero.
    v2f a; a.x = v;    a.y = 0.0f;
    v2f b; b.x = 1.0f; b.y = 1.0f;   // all-ones B: layout-independent
    v8f c = {};
    // D[m][n] = sum_k A[m][k] * 1 = v_m + v_{m+16}, replicated over n.
    v8f d = __builtin_amdgcn_wmma_f32_16x16x4_f32(
        /*neg_a=*/false, a, /*neg_b=*/false, b,
        /*c_mod=*/(short)0, c, /*reuse_a=*/false, /*reuse_b=*/false);

    // D layout: VGPR r holds M=r (lanes 0-15) / M=8+r (lanes 16-31).
    // Per-lane sum over the 8 VGPRs gives sum_{m=0..7} (lanes 0-15) or
    // sum_{m=8..15} (lanes 16-31); one xor-16 shuffle completes the total.
    float x = d[0] + d[1] + d[2] + d[3] + d[4] + d[5] + d[6] + d[7];
    float total = x + __shfl_xor(x, 16, 32);

    if (lane == 0) out[0] = total * inv_b;
}

extern "C" void kernel_launch(void* const* d_in, const int* in_sizes, int n_in,
                              void* d_out, int out_size, void* d_ws, size_t ws_size,
                              hipStream_t stream) {
    // Input order from setup_inputs(): hidden, logits, bias, teacher_probs, labels.
    // d_in[0] (hidden) is unused by the reference -> never read it.
    const float* logits  = (const float*)d_in[1];
    const float* bias    = (const float*)d_in[2];
    const float* teacher = (const float*)d_in[3];
    const int*   labels  = (const int*)  d_in[4];

    const int nrows = in_sizes[4];                 // B (131072)
    const int threads = 256;
    const int blocks = (nrows + threads - 1) / threads;  // 512 for B=131072

    float* partials = (float*)d_ws;                // blocks * 4 bytes scratch

    psd_loss_rows_kernel<<<blocks, threads, 0, stream>>>(
        logits, bias, teacher, labels, partials, nrows);

    psd_loss_reduce_kernel<<<1, 32, 0, stream>>>(
        partials, blocks, (float*)d_out, 1.0f / (float)nrows);
}